// QuantumHydraSSMCoreAdvanced_83863531421708
// MI455X (gfx1250) — compile-verified
//
#include <hip/hip_runtime.h>
#include <math.h>

#define NQ   8
#define DIM  256
#define QDEG 3
#define SEQ  256
#define PI_F 3.14159265358979f

typedef float v2f __attribute__((ext_vector_type(2)));
typedef float v8f __attribute__((ext_vector_type(8)));

// ---------------- Kernel 0: zero output (we accumulate fwd+bwd with atomics) ----
__global__ void k_zero(float* out, int n) {
  int i = blockIdx.x * blockDim.x + threadIdx.x;
  if (i < n) out[i] = 0.0f;
}

// ---------------- Kernel 1: build fixed middle unitary W (per direction) -------
// One block = one basis column of W for one param set. State in LDS, 256 threads,
// one amplitude per thread. Result stored in WMMA-A-tile layout:
//   element (M, K) of tile (mt, kt):  lane = M + 16*(K>>1), comp = K&1
//   float offset = ((set*2 + s)*16 + mt)*4096 + kt*64 + lane*2 + comp   (s: 0=re,1=im)

__device__ __forceinline__ void gate1q(float* pr, float* pi, int t, int q,
    float u00r, float u00i, float u01r, float u01i,
    float u10r, float u10i, float u11r, float u11i) {
  int mask = 1 << (7 - q);
  int b = t & mask;
  int p = t ^ mask;
  __syncthreads();
  float ar = pr[t], ai = pi[t], br = pr[p], bi = pi[p];
  float dr = b ? u11r : u00r, di = b ? u11i : u00i;
  float er = b ? u10r : u01r, ei = b ? u10i : u01i;
  float nr = dr*ar - di*ai + er*br - ei*bi;
  float ni = dr*ai + di*ar + er*bi + ei*br;
  __syncthreads();
  pr[t] = nr; pi[t] = ni;
}

// controlled gate whose target 2x2 is symmetric: diag=(dr,di), offdiag=(er,ei)
__device__ __forceinline__ void ctrl_sym(float* pr, float* pi, int t, int cq, int tq,
    float dr, float di, float er, float ei) {
  int cmask = 1 << (7 - cq);
  int tmask = 1 << (7 - tq);
  int p = t ^ tmask;
  __syncthreads();
  float ar = pr[t], ai = pi[t], br = pr[p], bi = pi[p];
  float nr = ar, ni = ai;
  if (t & cmask) {
    nr = dr*ar - di*ai + er*br - ei*bi;
    ni = dr*ai + di*ar + er*bi + ei*br;
  }
  __syncthreads();
  pr[t] = nr; pi[t] = ni;
}

__device__ __forceinline__ void rz_diag(float* pr, float* pi, int t, int q, float th) {
  int mask = 1 << (7 - q);
  float c = cosf(0.5f*th), s = sinf(0.5f*th);
  float ph = (t & mask) ? s : -s;            // e^{-i th/2} for bit=0, e^{+i th/2} for bit=1
  float ar = pr[t], ai = pi[t];
  pr[t] = c*ar - ph*ai;
  pi[t] = c*ai + ph*ar;
}

__global__ __launch_bounds__(256) void k_build_w(float* __restrict__ W,
                          const float* __restrict__ fwdp,
                          const float* __restrict__ bwdp,
                          const float* __restrict__ poly,
                          const float* __restrict__ qsvt) {
  __shared__ float pr[DIM];
  __shared__ float pim[DIM];
  int set = blockIdx.x >> 8;      // 0 = fwd params, 1 = bwd params
  int col = blockIdx.x & 255;     // input basis state (column of W)
  int t = threadIdx.x;
  const float* cp = set ? bwdp : fwdp;
  pr[t] = (t == col) ? 1.0f : 0.0f;
  pim[t] = 0.0f;
  // ---- QSVT section ----
  for (int deg = 0; deg <= QDEG; ++deg) {
    float coeff = poly[deg];
    for (int i = 0; i < NQ; ++i) {
      int idx = deg*NQ + i;
      float ang = (idx < NQ*QDEG) ? coeff*qsvt[idx]*PI_F : coeff*PI_F;
      rz_diag(pr, pim, t, i, ang);
    }
    for (int i = 0; i < NQ-1; ++i) ctrl_sym(pr, pim, t, i, i+1, 0.f,0.f, 1.f,0.f); // CNOT
    ctrl_sym(pr, pim, t, NQ-1, 0, 0.f,0.f, 1.f,0.f);
    for (int i = NQ-1; i >= 1; --i) ctrl_sym(pr, pim, t, i, i-1, 0.f,0.f, 1.f,0.f);
  }
  // ---- variational layers ----
  int p = 0;
  for (int l = 0; l < 2; ++l) {
    for (int i = 0; i < NQ; ++i) {
      float thx = cp[p], thy = cp[p+1], thz = cp[p+2]; p += 3;
      float c = cosf(0.5f*thx), s = sinf(0.5f*thx);
      gate1q(pr, pim, t, i, c,0.f, 0.f,-s, 0.f,-s, c,0.f);   // RX
      c = cosf(0.5f*thy); s = sinf(0.5f*thy);
      gate1q(pr, pim, t, i, c,0.f, -s,0.f, s,0.f, c,0.f);    // RY
      rz_diag(pr, pim, t, i, thz);                           // RZ
    }
    for (int i = 0; i < NQ; ++i) {
      float th = cp[p]; p += 1;
      float c = cosf(0.5f*th), s = sinf(0.5f*th);
      ctrl_sym(pr, pim, t, i, (i+1)&7, c,0.f, 0.f,-s);       // CRX(i -> i+1)
    }
    for (int i = NQ-1; i >= 0; --i) {
      float th = cp[p]; p += 1;
      float c = cosf(0.5f*th), s = sinf(0.5f*th);
      ctrl_sym(pr, pim, t, i, (i+7)&7, c,0.f, 0.f,-s);       // CRX(i -> i-1)
    }
  }
  __syncthreads();
  // store amplitude t of column col in WMMA A-tile layout
  int mt = t >> 4, M = t & 15;
  int kt = col >> 2, Kin = col & 3;
  int lane = M + ((Kin >> 1) << 4);
  int comp = Kin & 1;
  size_t base = (((size_t)set*2 + 0)*16 + mt)*4096 + (size_t)kt*64 + lane*2 + comp;
  W[base]            = pr[t];    // real part (s=0)
  W[base + 16*4096]  = pim[t];   // imag part (s=1)
}

// ---------------- Kernel 2: sequential SSM scan with per-step WMMA GEMM --------
// grid = 8: blockIdx = dir*4 + batch-group (16 samples each). 512 threads = 16 waves.
// Per step: psi[256][16] = W(dir) @ V,  V[k][n] = prod_q cs[n][q][bit],  then measure.
__global__ __launch_bounds__(512) void k_scan(const float* __restrict__ W,
                       const float* __restrict__ angles,
                       const float* __restrict__ fc,
                       const float* __restrict__ bc,
                       float* __restrict__ out) {
  __shared__ float V[DIM][16];      // real product-state matrix (K x N)
  __shared__ float ppr[DIM][16];    // psi real
  __shared__ float ppi[DIM][16];    // psi imag
  __shared__ float cs[16][NQ][2];
  __shared__ float hbuf[16][NQ];
  __shared__ float red[512][3];
  int tid  = threadIdx.x;
  int lane = tid & 31;
  int w    = tid >> 5;                   // wave id == m-tile (0..15)
  int dir  = blockIdx.x >> 2;
  int grp  = blockIdx.x & 3;
  float co  = dir ? bc[0] : fc[0];
  float sig = 1.0f / (1.0f + expf(-co));
  if (tid < 128) hbuf[tid >> 3][tid & 7] = 0.0f;
  const v2f* Wr = (const v2f*)W + (((size_t)dir*2 + 0)*16 + w)*2048;
  const v2f* Wi = (const v2f*)W + (((size_t)dir*2 + 1)*16 + w)*2048;
  int kn0 = lane >> 4;                   // B-tile K sublane (0/1)
  int nn  = lane & 15;                   // B/C column (local sample)
  __syncthreads();

  for (int it = 0; it < SEQ; ++it) {
    int tstep = dir ? (SEQ - 1 - it) : it;
    // 1) cos/sin of h/2
    if (tid < 128) {
      int n = tid >> 3, q = tid & 7;
      float hv = hbuf[n][q];
      cs[n][q][0] = cosf(0.5f * hv);
      cs[n][q][1] = sinf(0.5f * hv);
    }
    __syncthreads();
    // 2) real product state V[k][n] (qubit 0 = MSB of k)
    for (int r = 0; r < 8; ++r) {
      int idx = tid + 512*r;
      int k = idx >> 4, n = idx & 15;
      float v = 1.0f;
      #pragma unroll
      for (int q = 0; q < NQ; ++q) v *= cs[n][q][(k >> (7 - q)) & 1];
      V[k][n] = v;
    }
    __syncthreads();
    // 3) GEMM: this wave's 16-row tile of psi = W @ V  (2 real GEMMs: re, im)
    v8f cr = {0,0,0,0,0,0,0,0};
    v8f ci = {0,0,0,0,0,0,0,0};
    for (int kt = 0; kt < 64; ++kt) {
      v2f ar = Wr[kt*32 + lane];
      v2f ai = Wi[kt*32 + lane];
      v2f b;
      b.x = V[kt*4 + kn0][nn];
      b.y = V[kt*4 + 2 + kn0][nn];
      cr = __builtin_amdgcn_wmma_f32_16x16x4_f32(false, ar, false, b, (short)0, cr, false, false);
      ci = __builtin_amdgcn_wmma_f32_16x16x4_f32(false, ai, false, b, (short)0, ci, false, false);
    }
    // 4) spill psi tile to LDS (C layout: M = v + 8*(lane>>4), N = lane&15)
    #pragma unroll
    for (int v = 0; v < 8; ++v) {
      int M = v + ((lane >> 4) << 3);
      ppr[w*16 + M][nn] = cr[v];
      ppi[w*16 + M][nn] = ci[v];
    }
    __syncthreads();
    // 5) measurement partials: thread -> (qubit q, sample n, quarter part)
    {
      int q = tid >> 6, n = (tid >> 2) & 15, part = tid & 3;
      int mask = 1 << (7 - q);
      float sr = 0.f, si = 0.f, sz = 0.f;
      for (int jj = 0; jj < 32; ++jj) {
        int j = part*32 + jj;
        int low = j & (mask - 1);
        int i0 = ((j ^ low) << 1) | low;
        int i1 = i0 | mask;
        float r0 = ppr[i0][n], q0 = ppi[i0][n];
        float r1 = ppr[i1][n], q1 = ppi[i1][n];
        sr += r0*r1 + q0*q1;              // Re(conj(psi0)*psi1)
        si += r0*q1 - q0*r1;              // Im(conj(psi0)*psi1)
        sz += (r0*r0 + q0*q0) - (r1*r1 + q1*q1);
      }
      red[tid][0] = sr; red[tid][1] = si; red[tid][2] = sz;
    }
    __syncthreads();
    // 6) reduce + rotated-observable combine (absorbs the final RY(x) layer)
    if (tid < 128) {
      int q = tid >> 4, n = tid & 15;
      int base = tid << 2;
      float sr = red[base][0] + red[base+1][0] + red[base+2][0] + red[base+3][0];
      float si = red[base][1] + red[base+1][1] + red[base+2][1] + red[base+3][1];
      float sz = red[base][2] + red[base+1][2] + red[base+2][2] + red[base+3][2];
      float X = 2.0f*sr, Y = 2.0f*si, Z = sz;
      int b = grp*16 + n;
      float xa = angles[((size_t)b*SEQ + tstep)*NQ + q];
      float cx = cosf(xa), sx = sinf(xa);
      float Xo = cx*X + sx*Z;             // R'X R = cos X + sin Z
      float Zo = cx*Z - sx*X;             // R'Z R = cos Z - sin X
      size_t ob = ((size_t)b*SEQ + tstep)*24;
      atomicAdd(&out[ob + q],      sig*Xo);
      atomicAdd(&out[ob + 8 + q],  sig*Y);
      atomicAdd(&out[ob + 16 + q], sig*Zo);
      hbuf[n][q] = Zo;                    // h for next step
    }
    __syncthreads();
  }
}

extern "C" void kernel_launch(void* const* d_in, const int* in_sizes, int n_in,
                              void* d_out, int out_size, void* d_ws, size_t ws_size,
                              hipStream_t stream) {
  (void)in_sizes; (void)n_in; (void)ws_size;
  const float* angles = (const float*)d_in[0];
  const float* fwdp   = (const float*)d_in[1];
  const float* bwdp   = (const float*)d_in[2];
  const float* poly   = (const float*)d_in[3];
  const float* qsvt   = (const float*)d_in[4];
  const float* fc     = (const float*)d_in[5];
  const float* bc     = (const float*)d_in[6];
  float* out = (float*)d_out;
  float* W   = (float*)d_ws;   // 2 sets x (re,im) x 16 mt x 64 kt x 64 floats = 1 MB

  k_zero<<<(out_size + 511) / 512, 512, 0, stream>>>(out, out_size);
  k_build_w<<<512, 256, 0, stream>>>(W, fwdp, bwdp, poly, qsvt);
  k_scan<<<8, 512, 0, stream>>>(W, angles, fc, bc, out);
}